// RPEMultiheadAttention_4715874091072
// MI455X (gfx1250) — compile-verified
//
#include <hip/hip_runtime.h>
#include <math.h>

typedef __attribute__((ext_vector_type(2))) float v2f;
typedef __attribute__((ext_vector_type(4))) float v4f;
typedef __attribute__((ext_vector_type(8))) float v8f;

#define N_SEQ 512
#define M_SEQ 512
#define C_DIM 256
#define H_NUM 8
#define DH    32
#define K_EXT 512   // C (embed_g part) + C (key part)

// ---------------------------------------------------------------------------
// Kernel 1: q/k/v linear projections.  out[n,co] = x[n,:] . W[co,:] + b[co]
// ---------------------------------------------------------------------------
__global__ __launch_bounds__(256) void qkv_kernel(
    const float* __restrict__ query, const float* __restrict__ key,
    const float* __restrict__ value,
    const float* __restrict__ Wq, const float* __restrict__ bq,
    const float* __restrict__ Wk, const float* __restrict__ bk,
    const float* __restrict__ Wv, const float* __restrict__ bv,
    float* __restrict__ qo, float* __restrict__ ko, float* __restrict__ vo)
{
    int idx   = blockIdx.x * blockDim.x + threadIdx.x;   // [0, 3*512*256)
    int which = idx / (N_SEQ * C_DIM);
    int rem   = idx - which * (N_SEQ * C_DIM);
    int n     = rem >> 8;
    int co    = rem & (C_DIM - 1);

    const float *x, *W, *b;
    float* o;
    if (which == 0)      { x = query; W = Wq; b = bq; o = qo; }
    else if (which == 1) { x = key;   W = Wk; b = bk; o = ko; }
    else                 { x = value; W = Wv; b = bv; o = vo; }

    const v4f* xr = (const v4f*)(x + (size_t)n * C_DIM);
    const v4f* wr = (const v4f*)(W + (size_t)co * C_DIM);
    float acc = b[co];
#pragma unroll 8
    for (int i = 0; i < C_DIM / 4; ++i) {
        v4f a = xr[i], w = wr[i];
        acc += a.x * w.x + a.y * w.y + a.z * w.z + a.w * w.w;
    }
    o[(size_t)n * C_DIM + co] = acc;
}

// ---------------------------------------------------------------------------
// Kernel 2: one block per query row n (see round-0 analysis).
//   scores[h,m] = WMMA_{K=512}( [embed_g[n,m,:], k[m,:]] , [u[n,h,:], q_mask] )
//   embed_g is streamed once (256 MB > 192 MB L2)  -> non-temporal loads.
//   k/v/q/Wg are reused by all 512 blocks          -> default (RT) hints.
// ---------------------------------------------------------------------------
__global__ __launch_bounds__(256) void attn_kernel(
    const float* __restrict__ q, const float* __restrict__ kmat,
    const float* __restrict__ vmat, const float* __restrict__ embed_g,
    const float* __restrict__ Wg, const float* __restrict__ bg,
    float* __restrict__ out_hidden, float* __restrict__ out_attn)
{
    __shared__ float s_u[K_EXT][16];      // 32 KB  B-operand (heads padded to 16)
    __shared__ float s_sc[H_NUM][M_SEQ];  // 16 KB  scores -> probabilities
    __shared__ float s_qb[H_NUM];

    const int n   = blockIdx.x;
    const int tid = threadIdx.x;
    const float* qn = q + (size_t)n * C_DIM;

    // ---- stage B operand into LDS ----
    for (int idx = tid; idx < 256 * 16; idx += 256) {
        int kk = idx >> 4, h = idx & 15;
        float acc = 0.f;
        if (h < H_NUM) {
            int co0 = h * DH;
#pragma unroll 8
            for (int d = 0; d < DH; ++d)
                acc += qn[co0 + d] * Wg[(size_t)(co0 + d) * C_DIM + kk];
        }
        s_u[kk][h] = acc;
    }
    for (int idx = tid; idx < 256 * 16; idx += 256) {
        int co = idx >> 4, h = idx & 15;
        s_u[256 + co][h] = ((co >> 5) == h) ? qn[co] : 0.f;
    }
    if (tid < H_NUM) {
        float acc = 0.f;
        for (int d = 0; d < DH; ++d) acc += qn[tid * DH + d] * bg[tid * DH + d];
        s_qb[tid] = acc;
    }
    __syncthreads();

    // ---- WMMA score GEMM: 8 waves x 4 m-tiles each ----
    const int wave  = tid >> 5;
    const int lane  = tid & 31;
    const int mrow  = lane & 15;            // A-row in tile / B-col (head)
    const int khalf = (lane >> 4) << 1;     // lanes 16-31 hold K+2,K+3
    const int m0    = wave * 64;

    v8f acc0 = {}, acc1 = {}, acc2 = {}, acc3 = {};
    const float* Abase = embed_g + ((size_t)n * M_SEQ) * C_DIM;

    // K 0..255: embed_g contribution (streamed once -> non-temporal loads)
    for (int kc = 0; kc < 256; kc += 4) {
        int kb = kc + khalf;
        v2f B; B.x = s_u[kb][mrow]; B.y = s_u[kb + 1][mrow];
        v2f A0 = __builtin_nontemporal_load((const v2f*)(Abase + (size_t)(m0 +      mrow) * C_DIM + kb));
        v2f A1 = __builtin_nontemporal_load((const v2f*)(Abase + (size_t)(m0 + 16 + mrow) * C_DIM + kb));
        v2f A2 = __builtin_nontemporal_load((const v2f*)(Abase + (size_t)(m0 + 32 + mrow) * C_DIM + kb));
        v2f A3 = __builtin_nontemporal_load((const v2f*)(Abase + (size_t)(m0 + 48 + mrow) * C_DIM + kb));
        acc0 = __builtin_amdgcn_wmma_f32_16x16x4_f32(false, A0, false, B, (short)0, acc0, false, false);
        acc1 = __builtin_amdgcn_wmma_f32_16x16x4_f32(false, A1, false, B, (short)0, acc1, false, false);
        acc2 = __builtin_amdgcn_wmma_f32_16x16x4_f32(false, A2, false, B, (short)0, acc2, false, false);
        acc3 = __builtin_amdgcn_wmma_f32_16x16x4_f32(false, A3, false, B, (short)0, acc3, false, false);
    }
    // K 256..511: key contribution (k reused by all blocks -> regular loads)
    for (int kc = 0; kc < 256; kc += 4) {
        int kb = kc + khalf;
        v2f B; B.x = s_u[256 + kb][mrow]; B.y = s_u[256 + kb + 1][mrow];
        v2f A0 = *(const v2f*)(kmat + (size_t)(m0 +      mrow) * C_DIM + kb);
        v2f A1 = *(const v2f*)(kmat + (size_t)(m0 + 16 + mrow) * C_DIM + kb);
        v2f A2 = *(const v2f*)(kmat + (size_t)(m0 + 32 + mrow) * C_DIM + kb);
        v2f A3 = *(const v2f*)(kmat + (size_t)(m0 + 48 + mrow) * C_DIM + kb);
        acc0 = __builtin_amdgcn_wmma_f32_16x16x4_f32(false, A0, false, B, (short)0, acc0, false, false);
        acc1 = __builtin_amdgcn_wmma_f32_16x16x4_f32(false, A1, false, B, (short)0, acc1, false, false);
        acc2 = __builtin_amdgcn_wmma_f32_16x16x4_f32(false, A2, false, B, (short)0, acc2, false, false);
        acc3 = __builtin_amdgcn_wmma_f32_16x16x4_f32(false, A3, false, B, (short)0, acc3, false, false);
    }

    // ---- scatter D fragments (scale + bias) to LDS ----
    const float inv_sqrt = 0.17677669529663687f;   // 1/sqrt(32)
    if (mrow < H_NUM) {
        float qb = s_qb[mrow];
        int mbase = m0 + ((lane >> 4) << 3);       // D rows: j (+8 for hi lanes)
#pragma unroll
        for (int j = 0; j < 8; ++j) {
            s_sc[mrow][mbase + j]      = (acc0[j] + qb) * inv_sqrt;
            s_sc[mrow][mbase + 16 + j] = (acc1[j] + qb) * inv_sqrt;
            s_sc[mrow][mbase + 32 + j] = (acc2[j] + qb) * inv_sqrt;
            s_sc[mrow][mbase + 48 + j] = (acc3[j] + qb) * inv_sqrt;
        }
    }
    __syncthreads();

    // ---- softmax over m: wave w handles head w (wave32 reductions) ----
    {
        const int h = wave;
        float vals[16];
        float mx = -INFINITY;
#pragma unroll
        for (int i = 0; i < 16; ++i) {
            vals[i] = s_sc[h][lane + 32 * i];
            mx = fmaxf(mx, vals[i]);
        }
        for (int off = 16; off > 0; off >>= 1)
            mx = fmaxf(mx, __shfl_xor(mx, off, 32));
        float sum = 0.f;
#pragma unroll
        for (int i = 0; i < 16; ++i) {
            vals[i] = __expf(vals[i] - mx);
            sum += vals[i];
        }
        for (int off = 16; off > 0; off >>= 1)
            sum += __shfl_xor(sum, off, 32);
        float r = 1.0f / sum;
        float* attn_row = out_attn + ((size_t)h * N_SEQ + n) * M_SEQ;
#pragma unroll
        for (int i = 0; i < 16; ++i) {
            float p = vals[i] * r;
            s_sc[h][lane + 32 * i] = p;
            // attn is written once, never re-read on device -> NT store
            __builtin_nontemporal_store(p, attn_row + lane + 32 * i);
        }
    }
    __syncthreads();

    // ---- hidden = attn @ v  (thread = (h,d), coalesced v reads) ----
    {
        const int h = tid >> 5;
        const int d = tid & 31;
        float acc = 0.f;
        const float* vcol = vmat + h * DH + d;
        for (int m = 0; m < M_SEQ; m += 4) {
            acc += s_sc[h][m]     * vcol[(size_t)m * C_DIM];
            acc += s_sc[h][m + 1] * vcol[(size_t)(m + 1) * C_DIM];
            acc += s_sc[h][m + 2] * vcol[(size_t)(m + 2) * C_DIM];
            acc += s_sc[h][m + 3] * vcol[(size_t)(m + 3) * C_DIM];
        }
        out_hidden[(size_t)n * C_DIM + h * DH + d] = acc;
    }
}

// ---------------------------------------------------------------------------
extern "C" void kernel_launch(void* const* d_in, const int* in_sizes, int n_in,
                              void* d_out, int out_size, void* d_ws, size_t ws_size,
                              hipStream_t stream) {
    const float* query   = (const float*)d_in[0];
    const float* key     = (const float*)d_in[1];
    const float* value   = (const float*)d_in[2];
    const float* embed_g = (const float*)d_in[3];
    const float* Wq      = (const float*)d_in[4];
    const float* bq      = (const float*)d_in[5];
    const float* Wk      = (const float*)d_in[6];
    const float* bk      = (const float*)d_in[7];
    const float* Wv      = (const float*)d_in[8];
    const float* bv      = (const float*)d_in[9];
    const float* Wg      = (const float*)d_in[10];
    const float* bg      = (const float*)d_in[11];

    float* out        = (float*)d_out;
    float* out_hidden = out;                       // (N,B,C) = 512*256
    float* out_attn   = out + N_SEQ * C_DIM;       // (B,H,N,M) = 8*512*512

    float* qo = (float*)d_ws;                      // 512*256 each
    float* ko = qo + N_SEQ * C_DIM;
    float* vo = ko + M_SEQ * C_DIM;

    qkv_kernel<<<(3 * N_SEQ * C_DIM) / 256, 256, 0, stream>>>(
        query, key, value, Wq, bq, Wk, bk, Wv, bv, qo, ko, vo);
    attn_kernel<<<N_SEQ, 256, 0, stream>>>(
        qo, ko, vo, embed_g, Wg, bg, out_hidden, out_attn);
}